// LayoutBBox_77867757077175
// MI455X (gfx1250) — compile-verified
//
#include <hip/hip_runtime.h>

// Problem constants (from reference)
#define Wg   128
#define Hg   128
#define NUM  64
#define CLS  8
#define DIM  4

#define AS_GLOBAL __attribute__((address_space(1)))
#define AS_LDS    __attribute__((address_space(3)))

typedef int v4i __attribute__((ext_vector_type(4)));

#if defined(__has_builtin)
#  if __has_builtin(__builtin_amdgcn_global_load_async_to_lds_b128)
#    define HAVE_ASYNC_B128 1
#  endif
#  if __has_builtin(__builtin_amdgcn_s_wait_asynccnt)
#    define HAVE_WAIT_ASYNC 1
#  endif
#endif

// Async global->LDS copy of 16 bytes (CDNA5 ASYNCcnt path).
// Builtin signature (per hipcc diagnostic): (v4i AS1*, v4i AS3*, imm offset, imm cpol)
__device__ __forceinline__ void async_copy16(const float* g, float* l) {
#if defined(HAVE_ASYNC_B128)
  __builtin_amdgcn_global_load_async_to_lds_b128(
      (AS_GLOBAL v4i*)g, (AS_LDS v4i*)l, /*offset=*/0, /*cpol=*/0);
#else
  unsigned lds_off = (unsigned)(unsigned long long)((AS_LDS float*)l);
  asm volatile("global_load_async_to_lds_b128 %0, %1, off"
               :: "v"(lds_off), "v"(g)
               : "memory");
#endif
}

__device__ __forceinline__ void wait_async0() {
#if defined(HAVE_WAIT_ASYNC)
  __builtin_amdgcn_s_wait_asynccnt(0);
#else
  asm volatile("s_wait_asynccnt 0" ::: "memory");
#endif
}

__device__ __forceinline__ float clamp01(float v) {
  return fminf(fmaxf(v, 0.0f), 1.0f);
}

// One block per (batch, row). 128 threads = 4 wave32, thread = column index.
__global__ __launch_bounds__(128) void
layout_bbox_kernel(const float* __restrict__ in, float* __restrict__ out) {
  __shared__ float  s_in[(DIM + CLS) * NUM];  // raw (12,64) slice: 3 KB
  __shared__ float4 s_box[NUM];               // {x1, x2, fy(h), gy(h)} per box
  __shared__ float  s_cls[NUM * CLS];         // class probs transposed to [n][c]

  const int tid = threadIdx.x;
  const int b   = blockIdx.x >> 7;      // / Hg
  const int hh  = blockIdx.x & (Hg - 1);

  const float* gin = in + (size_t)b * (DIM + CLS) * NUM;
  __builtin_prefetch(gin, 0, 1);  // global_prefetch_b8

  // ---- Stage 768 floats (3 KB) into LDS via async DMA: 192 x b128 chunks.
  async_copy16(gin + 4 * tid, &s_in[4 * tid]);                    // chunks 0..127
  if (tid < 64)
    async_copy16(gin + 4 * (128 + tid), &s_in[4 * (128 + tid)]);  // chunks 128..191
  wait_async0();
  __syncthreads();

  // ---- Per-box precompute: x-edges + the h-dependent (row-constant) factors.
  if (tid < NUM) {
    const int n = tid;
    const float xc = s_in[0 * NUM + n] * (float)Wg;
    const float yc = s_in[1 * NUM + n] * (float)Hg;
    const float bw = s_in[2 * NUM + n] * (float)Wg;
    const float bh = s_in[3 * NUM + n] * (float)Hg;
    const float x1 = xc - 0.5f * bw, x2 = xc + 0.5f * bw;
    const float y1 = yc - 0.5f * bh, y2 = yc + 0.5f * bh;
    const float y1d = (float)hh - y1;
    const float y2d = y2 - (float)hh;
    const float gy  = clamp01(y1d) * clamp01(y2d);                     // "x_gate"
    const float fy  = fmaxf(fmaxf(1.0f - fabsf(y1d),
                                  1.0f - fabsf(y2d)), 0.0f);           // max y-line
    s_box[n] = make_float4(x1, x2, fy, gy);
#pragma unroll
    for (int c = 0; c < CLS; ++c)
      s_cls[n * CLS + c] = s_in[(DIM + c) * NUM + n];
  }
  __syncthreads();

  // ---- Main loop: thread owns one pixel (b, hh, ww=tid), 8-class running max.
  const float ww = (float)tid;
  float acc[CLS];
#pragma unroll
  for (int c = 0; c < CLS; ++c) acc[c] = 0.0f;

#pragma unroll 4
  for (int n = 0; n < NUM; ++n) {
    const float4 bx  = s_box[n];          // broadcast ds_load_b128
    const float  x1d = ww - bx.x;
    const float  x2d = bx.y - ww;
    const float  gx  = clamp01(x1d) * clamp01(x2d);                    // "y_gate"
    const float  fx  = fmaxf(fmaxf(1.0f - fabsf(x1d),
                                   1.0f - fabsf(x2d)), 0.0f);          // max x-line
    // xy_max = max(x-lines * y_gate-of-x, y-lines * x_gate-of-y)
    const float  xy  = fmaxf(fx * bx.w, bx.z * gx);
    if (xy > 0.0f) {  // most (pixel,box) pairs miss the box edges entirely
#pragma unroll
      for (int c = 0; c < CLS; ++c)
        acc[c] = fmaxf(acc[c], xy * s_cls[n * CLS + c]);
    }
  }

  // ---- Write (b, c, hh, ww); coalesced per class plane.
  float* po = out + (size_t)b * CLS * Hg * Wg + hh * Wg + tid;
#pragma unroll
  for (int c = 0; c < CLS; ++c) po[(size_t)c * Hg * Wg] = acc[c];
}

extern "C" void kernel_launch(void* const* d_in, const int* in_sizes, int n_in,
                              void* d_out, int out_size, void* d_ws, size_t ws_size,
                              hipStream_t stream) {
  (void)in_sizes; (void)n_in; (void)d_ws; (void)ws_size; (void)out_size;
  const float* in  = (const float*)d_in[0];
  float*       out = (float*)d_out;
  const int B = 16;
  layout_bbox_kernel<<<dim3(B * Hg), dim3(128), 0, stream>>>(in, out);
}